// Mamba3Layer_36653250904125
// MI455X (gfx1250) — compile-verified
//
#include <hip/hip_runtime.h>
#include <hip/hip_bf16.h>
#include <math.h>

// ---------------- problem constants (from reference) ----------------
#define BB      2
#define SS      4096
#define BSROWS  (BB*SS)        // 8192
#define DM      512            // D_MODEL
#define DI      1024           // D_INNER
#define NST     128            // D_STATE
#define NH      16             // heads
#define HP      64             // head dim
#define NPJ     2320           // d_in_proj = 2*DI + 2*NST + NH
#define XBCD    1280           // DI + 2*NST  (conv channels)
#define DCONV   4
#define NC      64             // chunks
#define TT      64             // tokens per chunk (Lg*r)
#define LG      16
#define RR      4

// ---------------- f32 WMMA (CDNA5: V_WMMA_F32_16X16X4_F32) ----------------
typedef float v2f_t __attribute__((ext_vector_type(2)));
typedef float v8f_t __attribute__((ext_vector_type(8)));

__device__ __forceinline__ v8f_t wmma4(float a0, float a1, float b0, float b1, v8f_t c) {
  v2f_t a; a[0] = a0; a[1] = a1;
  v2f_t b; b[0] = b0; b[1] = b1;
  return __builtin_amdgcn_wmma_f32_16x16x4_f32(false, a, false, b, (short)0, c, false, false);
}

// ======================================================================
// K1: zxbcdt[8192,2320] = (x*mask)[8192,512] @ W_in[512,2320]
// 32(M)x64(N) block tile, Kc=32, DOUBLE-BUFFERED LDS, b128 global loads.
// ======================================================================
__global__ __launch_bounds__(256) void k_inproj(const float* __restrict__ x,
                                                const unsigned char* __restrict__ mask,
                                                const float* __restrict__ Win,
                                                float* __restrict__ zx) {
  __shared__ float lA[2][32 * 33];
  __shared__ float lB[2][32 * 65];
  const int tid = threadIdx.x;
  const int lane = tid & 31, wv = tid >> 5;
  const int hl = lane >> 4, l16 = lane & 15;
  const int mblk = blockIdx.y * 32;
  const int nblk = blockIdx.x * 64;
  const int mbase = (wv >> 2) * 16, nbase = (wv & 3) * 16;

  // A: 32x32 floats = 256 float4 -> 1 per thread
  const int am = tid >> 3, ak = (tid & 7) * 4;
  // B: 32x64 floats = 512 float4 -> 2 per thread
  const int bk0 = tid >> 4, bk1 = (tid + 256) >> 4;
  const int bn = (tid & 15) * 4;
  const float ascale = mask[mblk + am] ? 1.0f : 0.0f;
  const int colb = nblk + bn;
  const bool bok = (colb < NPJ);

  float4 ra, rb0, rb1;
  auto gload = [&](int k0) {
    ra = *(const float4*)(x + (size_t)(mblk + am) * DM + k0 + ak);
    rb0 = bok ? *(const float4*)(Win + (size_t)(k0 + bk0) * NPJ + colb)
              : make_float4(0.f, 0.f, 0.f, 0.f);
    rb1 = bok ? *(const float4*)(Win + (size_t)(k0 + bk1) * NPJ + colb)
              : make_float4(0.f, 0.f, 0.f, 0.f);
  };
  auto sstore = [&](int buf) {
    float* a = &lA[buf][am * 33 + ak];
    a[0] = ra.x * ascale; a[1] = ra.y * ascale; a[2] = ra.z * ascale; a[3] = ra.w * ascale;
    float* p0 = &lB[buf][bk0 * 65 + bn];
    p0[0] = rb0.x; p0[1] = rb0.y; p0[2] = rb0.z; p0[3] = rb0.w;
    float* p1 = &lB[buf][bk1 * 65 + bn];
    p1[0] = rb1.x; p1[1] = rb1.y; p1[2] = rb1.z; p1[3] = rb1.w;
  };

  gload(0); sstore(0);
  __syncthreads();
  v8f_t acc = {};
  const int NSTG = DM / 32;
  for (int s = 0; s < NSTG; ++s) {
    const int cur = s & 1;
    if (s + 1 < NSTG) gload((s + 1) * 32);
#pragma unroll
    for (int kk = 0; kk < 32; kk += 4) {
      float a0 = lA[cur][(mbase + l16) * 33 + kk + 2 * hl];
      float a1 = lA[cur][(mbase + l16) * 33 + kk + 2 * hl + 1];
      float b0 = lB[cur][(kk + 2 * hl) * 65 + nbase + l16];
      float b1 = lB[cur][(kk + 2 * hl + 1) * 65 + nbase + l16];
      acc = wmma4(a0, a1, b0, b1, acc);
    }
    if (s + 1 < NSTG) sstore(cur ^ 1);
    __syncthreads();
  }
#pragma unroll
  for (int i = 0; i < 8; ++i) {
    int row = mblk + mbase + hl * 8 + i;
    int col = nblk + nbase + l16;
    if (col < NPJ) zx[(size_t)row * NPJ + col] = acc[i];
  }
}

// ======================================================================
// K2: causal depthwise conv (width 4) + SiLU over xbc channels
// ======================================================================
__global__ void k_conv(const float* __restrict__ zx, const float* __restrict__ cw,
                       const float* __restrict__ cbias, float* __restrict__ xbc) {
  int idx = blockIdx.x * 256 + threadIdx.x;
  if (idx >= BSROWS * XBCD) return;
  int row = idx / XBCD, c = idx % XBCD;
  int s = row & (SS - 1);
  float acc = cbias[c];
#pragma unroll
  for (int k = 0; k < DCONV; ++k) {
    int sp = s - (DCONV - 1) + k;
    if (sp >= 0) acc += cw[k * XBCD + c] * zx[(size_t)(row - (DCONV - 1) + k) * NPJ + DI + c];
  }
  xbc[idx] = acc / (1.0f + __expf(-acc));  // silu
}

// ======================================================================
// K3: dt = softplus(dt_raw + dt_bias)
// ======================================================================
__global__ void k_dtk(const float* __restrict__ zx, const float* __restrict__ dtb,
                      float* __restrict__ dtv) {
  int idx = blockIdx.x * 256 + threadIdx.x;
  if (idx >= BSROWS * NH) return;
  int row = idx >> 4, h = idx & 15;
  float v = zx[(size_t)row * NPJ + (2 * DI + 2 * NST) + h] + dtb[h];
  dtv[idx] = (v > 20.0f) ? v : log1pf(__expf(v));
}

// ======================================================================
// K4: per (b,c,h) cumulative decay: c_tok[64], c_last, chunk_dec
// ======================================================================
__global__ void k_cum(const float* __restrict__ dtv, const float* __restrict__ Alog,
                      float* __restrict__ ctok, float* __restrict__ clast,
                      float* __restrict__ cdec) {
  int tid = blockIdx.x * 256 + threadIdx.x;
  if (tid >= BB * NC * NH) return;
  int h = tid % NH, c = (tid / NH) % NC, b = tid / (NH * NC);
  float A = -__expf(Alog[h]);
  float run = 0.0f;
  int rowbase = b * SS + c * TT;
  for (int g = 0; g < LG; ++g) {
    float gs = 0.0f;
#pragma unroll
    for (int j = 0; j < RR; ++j) gs += dtv[(size_t)(rowbase + g * RR + j) * NH + h];
    run += gs * A;
#pragma unroll
    for (int j = 0; j < RR; ++j) ctok[(size_t)tid * TT + g * RR + j] = run;
  }
  clast[tid] = run;
  cdec[tid] = __expf(run);
}

// ======================================================================
// K5: scores[b,c] = Cc(64x128) @ Bc^T(128x64) via WMMA
// ======================================================================
__global__ __launch_bounds__(256) void k_scores(const float* __restrict__ xbc,
                                                float* __restrict__ scores) {
  __shared__ float lC[64 * 33];
  __shared__ float lBt[64 * 33];
  const int c = blockIdx.x, b = blockIdx.y;
  const int tid = threadIdx.x;
  const int lane = tid & 31, wv = tid >> 5;
  const int hl = lane >> 4, l16 = lane & 15;
  const int mb = (wv >> 2) * 16, nb = (wv & 3) * 16;
  const int rowbase = b * SS + c * TT;
  v8f_t acc0 = {}, acc1 = {};
  for (int n0 = 0; n0 < NST; n0 += 32) {
    // 64x32 floats = 512 float4 per array -> 2 float4 per thread each
    for (int e4 = tid; e4 < 512; e4 += 256) {
      int t = e4 >> 3, n = (e4 & 7) * 4;
      const float* rp = xbc + (size_t)(rowbase + t) * XBCD + DI;
      float4 vb = *(const float4*)(rp + n0 + n);
      float4 vc = *(const float4*)(rp + NST + n0 + n);
      float* pb = &lBt[t * 33 + n];
      pb[0] = vb.x; pb[1] = vb.y; pb[2] = vb.z; pb[3] = vb.w;
      float* pc = &lC[t * 33 + n];
      pc[0] = vc.x; pc[1] = vc.y; pc[2] = vc.z; pc[3] = vc.w;
    }
    __syncthreads();
#pragma unroll
    for (int kk = 0; kk < 32; kk += 4) {
      float a0 = lC[(mb + l16) * 33 + kk + 2 * hl];
      float a1 = lC[(mb + l16) * 33 + kk + 2 * hl + 1];
      float b0 = lBt[(nb + l16) * 33 + kk + 2 * hl];     // B^T operand
      float b1 = lBt[(nb + l16) * 33 + kk + 2 * hl + 1];
      acc0 = wmma4(a0, a1, b0, b1, acc0);
      float a2 = lC[(mb + 32 + l16) * 33 + kk + 2 * hl];
      float a3 = lC[(mb + 32 + l16) * 33 + kk + 2 * hl + 1];
      acc1 = wmma4(a2, a3, b0, b1, acc1);
    }
    __syncthreads();
  }
  float* sc = scores + (size_t)(b * NC + c) * TT * TT;
#pragma unroll
  for (int i = 0; i < 8; ++i) {
    int r = mb + hl * 8 + i, col = nb + l16;
    sc[r * TT + col] = acc0[i];
    sc[(r + 32) * TT + col] = acc1[i];
  }
}

// ======================================================================
// K6: per (b,c,h): y_intra = (seg*scores*decay) @ u  (+ D*x)
//                  states  = (B*dec_end)^T @ u
// ======================================================================
__global__ __launch_bounds__(256) void k_intra_states(
    const float* __restrict__ xbc, const float* __restrict__ dtv,
    const float* __restrict__ scores, const float* __restrict__ ctok,
    const float* __restrict__ clast, const float* __restrict__ Dp,
    float* __restrict__ y, float* __restrict__ states) {
  __shared__ float su[64 * 64];    // u[t][p]
  __shared__ float s2[64 * 128];   // first M[t][k] (64x64), then Bd[t][n] (64x128)
  __shared__ float sct[64];
  const int h = blockIdx.x, c = blockIdx.y, b = blockIdx.z;
  const int tid = threadIdx.x;
  const int lane = tid & 31, wv = tid >> 5;
  const int hl = lane >> 4, l16 = lane & 15;
  const int rowbase = b * SS + c * TT;
  const int cb = (b * NC + c) * NH + h;

  if (tid < 64) sct[tid] = ctok[(size_t)cb * TT + tid];
  __syncthreads();

  // u[t][p] = dt[t] * x[t][p]  (b128 loads, p-contiguous)
  for (int e4 = tid; e4 < 1024; e4 += 256) {
    int t = e4 >> 4, p = (e4 & 15) * 4;
    float dtt = dtv[(size_t)(rowbase + t) * NH + h];
    float4 v = *(const float4*)(xbc + (size_t)(rowbase + t) * XBCD + h * HP + p);
    float* pu = &su[t * 64 + p];
    pu[0] = dtt * v.x; pu[1] = dtt * v.y; pu[2] = dtt * v.z; pu[3] = dtt * v.w;
  }
  // M[t][k] = seg * scores * exp(ctok[t]-ctok[k])
  const float* sc = scores + (size_t)(b * NC + c) * TT * TT;
  for (int e = tid; e < 64 * 64; e += 256) {
    int t = e >> 6, k = e & 63;
    float m = 0.0f;
    if ((t >> 2) >= (k >> 2)) m = sc[t * TT + k] * __expf(sct[t] - sct[k]);
    s2[e] = m;
  }
  __syncthreads();

  // ---- y_intra: 64x64 = M @ u, 16 tiles, 2 per wave ----
  const int mb = (wv >> 2) * 16, nb = (wv & 3) * 16;
  v8f_t acc0 = {}, acc1 = {};
#pragma unroll
  for (int kk = 0; kk < 64; kk += 4) {
    float b0 = su[(kk + 2 * hl) * 64 + nb + l16];
    float b1 = su[(kk + 2 * hl + 1) * 64 + nb + l16];
    float a0 = s2[(mb + l16) * 64 + kk + 2 * hl];
    float a1 = s2[(mb + l16) * 64 + kk + 2 * hl + 1];
    acc0 = wmma4(a0, a1, b0, b1, acc0);
    float a2 = s2[(mb + 32 + l16) * 64 + kk + 2 * hl];
    float a3 = s2[(mb + 32 + l16) * 64 + kk + 2 * hl + 1];
    acc1 = wmma4(a2, a3, b0, b1, acc1);
  }
  float dparam = Dp[h];
#pragma unroll
  for (int i = 0; i < 8; ++i) {
    int t0 = mb + hl * 8 + i, col = nb + l16;
    size_t r0 = (size_t)(rowbase + t0);
    y[r0 * DI + h * HP + col] = acc0[i] + dparam * xbc[r0 * XBCD + h * HP + col];
    size_t r1 = (size_t)(rowbase + t0 + 32);
    y[r1 * DI + h * HP + col] = acc1[i] + dparam * xbc[r1 * XBCD + h * HP + col];
  }
  __syncthreads();

  // ---- build decayed B: Bd[t][n] = B[t,n]*exp(c_last - c_tok[t]) ----
  float cl = clast[cb];
  for (int e4 = tid; e4 < 2048; e4 += 256) {
    int t = e4 >> 5, n = (e4 & 31) * 4;
    float de = __expf(cl - sct[t]);
    float4 v = *(const float4*)(xbc + (size_t)(rowbase + t) * XBCD + DI + n);
    float* pb = &s2[t * 128 + n];
    pb[0] = de * v.x; pb[1] = de * v.y; pb[2] = de * v.z; pb[3] = de * v.w;
  }
  __syncthreads();

  // ---- states: 128x64 = Bd^T @ u, 32 tiles, 4 per wave ----
  float* st = states + (size_t)cb * NST * HP;
#pragma unroll
  for (int ti = 0; ti < 4; ++ti) {
    int tile = wv + ti * 8;
    int smb = (tile >> 2) * 16, snb = (tile & 3) * 16;
    v8f_t acc = {};
#pragma unroll
    for (int kk = 0; kk < 64; kk += 4) {
      float a0 = s2[(kk + 2 * hl) * 128 + smb + l16];      // Bd^T operand
      float a1 = s2[(kk + 2 * hl + 1) * 128 + smb + l16];
      float b0 = su[(kk + 2 * hl) * 64 + snb + l16];
      float b1 = su[(kk + 2 * hl + 1) * 64 + snb + l16];
      acc = wmma4(a0, a1, b0, b1, acc);
    }
#pragma unroll
    for (int i = 0; i < 8; ++i)
      st[(smb + hl * 8 + i) * HP + snb + l16] = acc[i];
  }
}

// ======================================================================
// K7: per (b,h): sequential scan over 64 chunks with h-state in LDS,
//     y_inter = exp(c_tok) * (Cc @ h_prev) accumulated into y
// ======================================================================
__global__ __launch_bounds__(256) void k_scan_inter(
    const float* __restrict__ xbc, const float* __restrict__ ctok,
    const float* __restrict__ cdec, const float* __restrict__ states,
    float* __restrict__ y) {
  __shared__ float sh[128 * 64];  // h[n][p]
  __shared__ float sC[64 * 64];   // Cc stage [t][nn]
  const int h = blockIdx.x, b = blockIdx.y;
  const int tid = threadIdx.x;
  const int lane = tid & 31, wv = tid >> 5;
  const int hl = lane >> 4, l16 = lane & 15;
  const int mb = (wv >> 2) * 16, nb = (wv & 3) * 16;

  for (int e = tid; e < 128 * 64; e += 256) sh[e] = 0.0f;
  __syncthreads();

  for (int c = 0; c < NC; ++c) {
    const int rowbase = b * SS + c * TT;
    const int cb = (b * NC + c) * NH + h;
    v8f_t acc0 = {}, acc1 = {};
    for (int n0 = 0; n0 < NST; n0 += 64) {
      // 64x64 floats = 1024 float4 -> 4 per thread
      for (int e4 = tid; e4 < 1024; e4 += 256) {
        int t = e4 >> 4, nn = (e4 & 15) * 4;
        float4 v = *(const float4*)(xbc + (size_t)(rowbase + t) * XBCD + DI + NST + n0 + nn);
        float* pc = &sC[t * 64 + nn];
        pc[0] = v.x; pc[1] = v.y; pc[2] = v.z; pc[3] = v.w;
      }
      __syncthreads();
#pragma unroll
      for (int kk = 0; kk < 64; kk += 4) {
        float a0 = sC[(mb + l16) * 64 + kk + 2 * hl];
        float a1 = sC[(mb + l16) * 64 + kk + 2 * hl + 1];
        float b0 = sh[(n0 + kk + 2 * hl) * 64 + nb + l16];
        float b1 = sh[(n0 + kk + 2 * hl + 1) * 64 + nb + l16];
        acc0 = wmma4(a0, a1, b0, b1, acc0);
        float a2 = sC[(mb + 32 + l16) * 64 + kk + 2 * hl];
        float a3 = sC[(mb + 32 + l16) * 64 + kk + 2 * hl + 1];
        acc1 = wmma4(a2, a3, b0, b1, acc1);
      }
      __syncthreads();
    }
#pragma unroll
    for (int i = 0; i < 8; ++i) {
      int t0 = mb + hl * 8 + i, col = nb + l16;
      float e0 = __expf(ctok[(size_t)cb * TT + t0]);
      y[(size_t)(rowbase + t0) * DI + h * HP + col] += e0 * acc0[i];
      float e1 = __expf(ctok[(size_t)cb * TT + t0 + 32]);
      y[(size_t)(rowbase + t0 + 32) * DI + h * HP + col] += e1 * acc1[i];
    }
    // ---- h = dec * h + states[b,c,h] ----
    float dec = cdec[cb];
    const float* st = states + (size_t)cb * NST * HP;
    for (int e = tid; e < 128 * 64; e += 256)
      sh[e] = dec * sh[e] + st[e];
    __syncthreads();
  }
}

// ======================================================================
// K8: y *= silu(z)
// ======================================================================
__global__ void k_gate(const float* __restrict__ zx, float* __restrict__ y) {
  int idx = blockIdx.x * 256 + threadIdx.x;
  if (idx >= BSROWS * DI) return;
  int row = idx >> 10;
  int d = idx & 1023;
  float z = zx[(size_t)row * NPJ + d];
  y[idx] *= z / (1.0f + __expf(-z));
}

// ======================================================================
// K9: out[8192,512] = y[8192,1024] @ W_out[1024,512], masked
// double-buffered LDS, b128 global loads
// ======================================================================
__global__ __launch_bounds__(256) void k_outproj(const float* __restrict__ y,
                                                 const float* __restrict__ Wout,
                                                 const unsigned char* __restrict__ mask,
                                                 float* __restrict__ out) {
  __shared__ float lA[2][32 * 33];
  __shared__ float lB[2][32 * 65];
  const int tid = threadIdx.x;
  const int lane = tid & 31, wv = tid >> 5;
  const int hl = lane >> 4, l16 = lane & 15;
  const int mblk = blockIdx.y * 32;
  const int nblk = blockIdx.x * 64;
  const int mbase = (wv >> 2) * 16, nbase = (wv & 3) * 16;

  const int am = tid >> 3, ak = (tid & 7) * 4;
  const int bk0 = tid >> 4, bk1 = (tid + 256) >> 4;
  const int bn = (tid & 15) * 4;
  const int colb = nblk + bn;

  float4 ra, rb0, rb1;
  auto gload = [&](int k0) {
    ra  = *(const float4*)(y + (size_t)(mblk + am) * DI + k0 + ak);
    rb0 = *(const float4*)(Wout + (size_t)(k0 + bk0) * DM + colb);
    rb1 = *(const float4*)(Wout + (size_t)(k0 + bk1) * DM + colb);
  };
  auto sstore = [&](int buf) {
    float* a = &lA[buf][am * 33 + ak];
    a[0] = ra.x; a[1] = ra.y; a[2] = ra.z; a[3] = ra.w;
    float* p0 = &lB[buf][bk0 * 65 + bn];
    p0[0] = rb0.x; p0[1] = rb0.y; p0[2] = rb0.z; p0[3] = rb0.w;
    float* p1 = &lB[buf][bk1 * 65 + bn];
    p1[0] = rb1.x; p1[1] = rb1.y; p1[2] = rb1.z; p1[3] = rb1.w;
  };

  gload(0); sstore(0);
  __syncthreads();
  v8f_t acc = {};
  const int NSTG = DI / 32;
  for (int s = 0; s < NSTG; ++s) {
    const int cur = s & 1;
    if (s + 1 < NSTG) gload((s + 1) * 32);
#pragma unroll
    for (int kk = 0; kk < 32; kk += 4) {
      float a0 = lA[cur][(mbase + l16) * 33 + kk + 2 * hl];
      float a1 = lA[cur][(mbase + l16) * 33 + kk + 2 * hl + 1];
      float b0 = lB[cur][(kk + 2 * hl) * 65 + nbase + l16];
      float b1 = lB[cur][(kk + 2 * hl + 1) * 65 + nbase + l16];
      acc = wmma4(a0, a1, b0, b1, acc);
    }
    if (s + 1 < NSTG) sstore(cur ^ 1);
    __syncthreads();
  }
#pragma unroll
  for (int i = 0; i < 8; ++i) {
    int row = mblk + mbase + hl * 8 + i;
    int col = nblk + nbase + l16;
    out[(size_t)row * DM + col] = mask[row] ? acc[i] : 0.0f;
  }
}

// ======================================================================
// workspace layout (floats); total ~211.5 MiB
// ======================================================================
static constexpr size_t OFF_ZX = 0;
static constexpr size_t SZ_ZX  = (size_t)BSROWS * NPJ;
static constexpr size_t OFF_XBC = OFF_ZX + SZ_ZX;
static constexpr size_t SZ_XBC  = (size_t)BSROWS * XBCD;
static constexpr size_t OFF_DT  = OFF_XBC + SZ_XBC;
static constexpr size_t SZ_DT   = (size_t)BSROWS * NH;
static constexpr size_t OFF_CT  = OFF_DT + SZ_DT;
static constexpr size_t SZ_CT   = (size_t)BB * NC * NH * TT;
static constexpr size_t OFF_CL  = OFF_CT + SZ_CT;
static constexpr size_t SZ_CL   = (size_t)BB * NC * NH;
static constexpr size_t OFF_CD  = OFF_CL + SZ_CL;
static constexpr size_t OFF_SC  = OFF_CD + SZ_CL;
static constexpr size_t SZ_SC   = (size_t)BB * NC * TT * TT;
static constexpr size_t OFF_ST  = OFF_SC + SZ_SC;
static constexpr size_t SZ_ST   = (size_t)BB * NC * NH * NST * HP;
static constexpr size_t OFF_Y   = OFF_ST + SZ_ST;

extern "C" void kernel_launch(void* const* d_in, const int* in_sizes, int n_in,
                              void* d_out, int out_size, void* d_ws, size_t ws_size,
                              hipStream_t stream) {
  const float*         x     = (const float*)d_in[0];
  const unsigned char* mask  = (const unsigned char*)d_in[1];
  const float*         Win   = (const float*)d_in[2];
  const float*         convw = (const float*)d_in[3];
  const float*         convb = (const float*)d_in[4];
  const float*         dtb   = (const float*)d_in[5];
  const float*         Alog  = (const float*)d_in[6];
  const float*         Dp    = (const float*)d_in[7];
  const float*         Wout  = (const float*)d_in[8];
  float*               out   = (float*)d_out;

  float* ws    = (float*)d_ws;
  float* zx    = ws + OFF_ZX;
  float* xbc   = ws + OFF_XBC;
  float* dtv   = ws + OFF_DT;
  float* ctok  = ws + OFF_CT;
  float* clast = ws + OFF_CL;
  float* cdec  = ws + OFF_CD;
  float* sc    = ws + OFF_SC;
  float* st    = ws + OFF_ST;
  float* y     = ws + OFF_Y;

  // 1. in-projection GEMM (WMMA f32, double-buffered)
  k_inproj<<<dim3((NPJ + 63) / 64, BSROWS / 32), 256, 0, stream>>>(x, mask, Win, zx);
  // 2. conv + silu
  k_conv<<<(BSROWS * XBCD + 255) / 256, 256, 0, stream>>>(zx, convw, convb, xbc);
  // 3. dt = softplus
  k_dtk<<<(BSROWS * NH + 255) / 256, 256, 0, stream>>>(zx, dtb, dtv);
  // 4. cumulative decay
  k_cum<<<(BB * NC * NH + 255) / 256, 256, 0, stream>>>(dtv, Alog, ctok, clast, cdec);
  // 5. scores (WMMA)
  k_scores<<<dim3(NC, BB), 256, 0, stream>>>(xbc, sc);
  // 6. intra-chunk output + chunk states (WMMA)
  k_intra_states<<<dim3(NH, NC, BB), 256, 0, stream>>>(xbc, dtv, sc, ctok, clast, Dp, y, st);
  // 7. sequential chunk scan + inter-chunk output (WMMA, h-state in LDS)
  k_scan_inter<<<dim3(NH, BB), 256, 0, stream>>>(xbc, ctok, cdec, st, y);
  // 8. silu gate
  k_gate<<<(BSROWS * DI + 255) / 256, 256, 0, stream>>>(zx, y);
  // 9. out-projection GEMM (WMMA, double-buffered)
  k_outproj<<<dim3(DM / 64, BSROWS / 32), 256, 0, stream>>>(y, Wout, mask, out);
}